// ProposalLayer_26130581028991
// MI455X (gfx1250) — compile-verified
//
#include <hip/hip_runtime.h>
#include <hip/hip_bf16.h>
#include <stdint.h>

// ---------------- Problem constants (fixed by setup_inputs) ----------------
#define FEAT_H   64
#define FEAT_W   100
#define NANCH    9
#define NBOX     (FEAT_H * FEAT_W * NANCH)   // 57600 anchors
#define SORT_N   65536                        // next pow2 >= NBOX
#define NTOP     6000                         // PRE_NMS
#define NOUT     300                          // POST_NMS
#define NWORD    188                          // ceil(NTOP/32)
#define NMS_TH   0.7f
#define NEG_SCORE -1e30f

#define CHUNK        16384                    // LDS-resident sort chunk: 128 KB of 320 KB LDS
#define SORT_THREADS 1024                     // 32 wave32 per block

// Exact base-anchor widths/heights (np.round semantics folded in at build time):
// ratios (0.5,1,2) x scales (8,16,32):
//   r=0.5: w=23,h=12 ; r=1: w=16,h=16 ; r=2: w=11,h=23
__constant__ float c_aw[NANCH] = {184.f, 368.f, 736.f, 128.f, 256.f, 512.f,  88.f, 176.f, 352.f};
__constant__ float c_ah[NANCH] = { 96.f, 192.f, 384.f, 128.f, 256.f, 512.f, 184.f, 368.f, 736.f};

// Map fp32 to a totally-ordered uint32 (descending sort == descending score).
__device__ __forceinline__ uint32_t f32_sortable(float f) {
    uint32_t b = __float_as_uint(f);
    return (b & 0x80000000u) ? ~b : (b | 0x80000000u);
}

// ---------------- K1: decode anchors -> proposals + sort keys ----------------
// key = sortable(masked_score) << 32 | ~index  (descending key sort == top_k order,
// ties broken toward lower index, exactly like jax.lax.top_k).
__global__ void decode_kernel(const float* __restrict__ cls,    // (18,64,100)
                              const float* __restrict__ bbox,   // (36,64,100)
                              const int* __restrict__ img_h_p,
                              const int* __restrict__ img_w_p,
                              float4* __restrict__ proposals,   // [NBOX]
                              unsigned long long* __restrict__ keys) { // [SORT_N]
    const int n = blockIdx.x * blockDim.x + threadIdx.x;
    if (n >= SORT_N) return;
    if (n >= NBOX) { keys[n] = 0ull; return; }  // pad sinks to bottom of descending sort

    const int HW = FEAT_H * FEAT_W;
    const int a    = n % NANCH;
    const int cell = n / NANCH;
    const int x    = cell % FEAT_W;
    const int y    = cell / FEAT_W;
    const int p    = y * FEAT_W + x;

    // softmax over 2 classes -> fg prob (== sigmoid(s1 - s0))
    const float s0 = cls[(2 * a + 0) * HW + p];
    const float s1 = cls[(2 * a + 1) * HW + p];
    const float score = 1.0f / (1.0f + expf(s0 - s1));

    const float dx = bbox[(4 * a + 0) * HW + p];
    const float dy = bbox[(4 * a + 1) * HW + p];
    const float dw = bbox[(4 * a + 2) * HW + p];
    const float dh = bbox[(4 * a + 3) * HW + p];

    const float aw = c_aw[a], ah = c_ah[a];
    // exact in fp32: width=aw, height=ah, center = 16*x + 8
    const float cx = 16.0f * (float)x + 8.0f;
    const float cy = 16.0f * (float)y + 8.0f;

    const float pcx = dx * aw + cx;
    const float pcy = dy * ah + cy;
    const float pw  = expf(dw) * aw;
    const float ph  = expf(dh) * ah;

    const float fw = (float)(*img_w_p) - 1.0f;
    const float fh = (float)(*img_h_p) - 1.0f;

    const float x1 = fminf(fmaxf(pcx - 0.5f * pw, 0.0f), fw);
    const float y1 = fminf(fmaxf(pcy - 0.5f * ph, 0.0f), fh);
    const float x2 = fminf(fmaxf(pcx + 0.5f * pw, 0.0f), fw);
    const float y2 = fminf(fmaxf(pcy + 0.5f * ph, 0.0f), fh);

    proposals[n] = make_float4(x1, y1, x2, y2);

    const bool valid = (x2 - x1 + 1.0f >= 16.0f) && (y2 - y1 + 1.0f >= 16.0f);
    const float masked = valid ? score : NEG_SCORE;

    const unsigned long long key =
        ((unsigned long long)f32_sortable(masked) << 32) | (uint32_t)(~(uint32_t)n);
    keys[n] = key;
}

// ---------------- Sort helpers ----------------
// Compare-exchange for DESCENDING overall order: in "up" subsequences larger-first.
__device__ __forceinline__ void cmpxchg_desc(unsigned long long& a, unsigned long long& b, bool up) {
    if (up ? (a < b) : (a > b)) {
        unsigned long long t = a; a = b; b = t;
    }
}

// ---------------- K2a: LDS-resident bitonic sort of one 16K chunk ----------------
// Executes all network stages k=2..CHUNK entirely in LDS (105 passes, barrier-synced).
// Chunk staged in/out with CDNA5 async global<->LDS ops (ASYNCcnt).
__global__ void __launch_bounds__(SORT_THREADS)
local_sort_kernel(unsigned long long* __restrict__ keys) {
    __shared__ unsigned long long sk[CHUNK];   // 128 KB (only shared object -> LDS offset 0)
    const int base = blockIdx.x * CHUNK;

    for (int l = threadIdx.x; l < CHUNK; l += SORT_THREADS) {
        const uint32_t loff = (uint32_t)(uintptr_t)(&sk[l]);
        const uint64_t g    = (uint64_t)(uintptr_t)(keys + base + l);
        asm volatile("global_load_async_to_lds_b64 %0, %1, off" :: "v"(loff), "v"(g) : "memory");
    }
    asm volatile("s_wait_asynccnt 0x0" ::: "memory");
    __syncthreads();

    for (int k = 2; k <= CHUNK; k <<= 1) {
        for (int j = k >> 1; j > 0; j >>= 1) {
            for (int p = threadIdx.x; p < CHUNK / 2; p += SORT_THREADS) {
                const int l = ((p & ~(j - 1)) << 1) | (p & (j - 1));  // index with bit j clear
                const int m = l | j;
                unsigned long long a = sk[l], b = sk[m];
                const bool up = (((base + l) & k) == 0);
                cmpxchg_desc(a, b, up);
                sk[l] = a; sk[m] = b;
            }
            __syncthreads();
        }
    }

    for (int l = threadIdx.x; l < CHUNK; l += SORT_THREADS) {
        const uint32_t loff = (uint32_t)(uintptr_t)(&sk[l]);
        const uint64_t g    = (uint64_t)(uintptr_t)(keys + base + l);
        asm volatile("global_store_async_from_lds_b64 %1, %0, off" :: "v"(loff), "v"(g) : "memory");
    }
    asm volatile("s_wait_asynccnt 0x0" ::: "memory");
}

// ---------------- K2b: LDS-resident merge: stage k, passes j=CHUNK/2..1 ----------------
__global__ void __launch_bounds__(SORT_THREADS)
local_merge_kernel(unsigned long long* __restrict__ keys, int k) {
    __shared__ unsigned long long sk[CHUNK];   // 128 KB
    const int base = blockIdx.x * CHUNK;

    for (int l = threadIdx.x; l < CHUNK; l += SORT_THREADS) {
        const uint32_t loff = (uint32_t)(uintptr_t)(&sk[l]);
        const uint64_t g    = (uint64_t)(uintptr_t)(keys + base + l);
        asm volatile("global_load_async_to_lds_b64 %0, %1, off" :: "v"(loff), "v"(g) : "memory");
    }
    asm volatile("s_wait_asynccnt 0x0" ::: "memory");
    __syncthreads();

    for (int j = CHUNK >> 1; j > 0; j >>= 1) {
        for (int p = threadIdx.x; p < CHUNK / 2; p += SORT_THREADS) {
            const int l = ((p & ~(j - 1)) << 1) | (p & (j - 1));
            const int m = l | j;
            unsigned long long a = sk[l], b = sk[m];
            const bool up = (((base + l) & k) == 0);
            cmpxchg_desc(a, b, up);
            sk[l] = a; sk[m] = b;
        }
        __syncthreads();
    }

    for (int l = threadIdx.x; l < CHUNK; l += SORT_THREADS) {
        const uint32_t loff = (uint32_t)(uintptr_t)(&sk[l]);
        const uint64_t g    = (uint64_t)(uintptr_t)(keys + base + l);
        asm volatile("global_store_async_from_lds_b64 %1, %0, off" :: "v"(loff), "v"(g) : "memory");
    }
    asm volatile("s_wait_asynccnt 0x0" ::: "memory");
}

// ---------------- K2c: one global compare-exchange pass (j >= CHUNK only) ----------------
__global__ void bitonic_pass_kernel(unsigned long long* __restrict__ keys, int k, int j) {
    const int t = blockIdx.x * blockDim.x + threadIdx.x;
    const int ixj = t ^ j;
    if (ixj > t) {
        const unsigned long long a = keys[t];
        const unsigned long long b = keys[ixj];
        const bool up = ((t & k) == 0);
        if (up ? (a < b) : (a > b)) {
            keys[t] = b;
            keys[ixj] = a;
        }
    }
}

// ---------------- K3: gather top-6000 boxes in score order ----------------
__global__ void gather_kernel(const unsigned long long* __restrict__ keys,
                              const float4* __restrict__ proposals,
                              float4* __restrict__ boxes) {
    const int t = blockIdx.x * blockDim.x + threadIdx.x;
    if (t >= NTOP) return;
    const uint32_t n = ~(uint32_t)(keys[t]);   // low 32 bits stored ~index
    boxes[t] = proposals[n];
}

// ---------------- K4: NMS suppression bitmask (parallel O(n^2) part) ----------------
// Block handles 64 consecutive i rows; j space is staged through LDS in tiles of
// 1024 boxes using CDNA5 async global->LDS loads (ASYNCcnt / s_wait_asynccnt).
#define JTILE 1024
__global__ void nms_mask_kernel(const float4* __restrict__ boxes,
                                unsigned* __restrict__ mask) {  // [NTOP][NWORD]
    __shared__ float4 sj[JTILE];     // only shared object -> LDS offset 0
    const int tid   = threadIdx.x;   // 256 threads = 8 wave32
    const int ibase = blockIdx.x * 64;

    for (int jt = 0; jt < (NTOP + JTILE - 1) / JTILE; ++jt) {
        const int j0 = jt * JTILE;
        // --- async-copy this j-tile of boxes into LDS (gfx1250 path) ---
        for (int t = tid; t < JTILE; t += 256) {
            const int j = j0 + t;
            if (j < NTOP) {
                const uint32_t loff  = (uint32_t)(uintptr_t)(&sj[t]);            // LDS byte addr
                const uint64_t gaddr = (uint64_t)(uintptr_t)(boxes + j);         // global addr
                asm volatile("global_load_async_to_lds_b128 %0, %1, off"
                             :: "v"(loff), "v"(gaddr) : "memory");
            }
        }
        asm volatile("s_wait_asynccnt 0x0" ::: "memory");
        __syncthreads();

        // --- compute: thread -> (i_local stride 8, word-in-tile = tid&31) ---
        const int wl = tid & 31;                 // 32 mask words per 1024-j tile
        const int w  = jt * 32 + wl;             // global word index
        for (int il = tid >> 5; il < 64; il += 8) {
            const int i = ibase + il;
            if (i >= NTOP || w >= NWORD) continue;
            const float4 bi = boxes[i];          // broadcast, L0-resident
            const float areai = (bi.z - bi.x) * (bi.w - bi.y);
            unsigned bits = 0u;
            const int jb = j0 + wl * 32;
            #pragma unroll
            for (int b = 0; b < 32; ++b) {
                const int j = jb + b;
                if (j < NTOP && j > i) {
                    const float4 bj = sj[wl * 32 + b];
                    const float lx = fmaxf(bi.x, bj.x);
                    const float ly = fmaxf(bi.y, bj.y);
                    const float rx = fminf(bi.z, bj.z);
                    const float ry = fminf(bi.w, bj.w);
                    const float iw = fmaxf(rx - lx, 0.0f);
                    const float ih = fmaxf(ry - ly, 0.0f);
                    const float inter = iw * ih;
                    const float areaj = (bj.z - bj.x) * (bj.w - bj.y);
                    const float iou = inter / (areai + areaj - inter);
                    if (iou > NMS_TH) bits |= (1u << b);
                }
            }
            mask[(size_t)i * NWORD + w] = bits;
        }
        __syncthreads();   // protect sj before next tile's async writes
    }
}

// ---------------- K5: sequential greedy sweep + emit (300,5) ----------------
__global__ void nms_sweep_kernel(const float4* __restrict__ boxes,
                                 const unsigned* __restrict__ mask,
                                 float* __restrict__ out) {
    __shared__ unsigned rem[NWORD];
    __shared__ int keep[NOUT];
    __shared__ int cnt;
    const int tid = threadIdx.x;   // 256 threads
    for (int w = tid; w < NWORD; w += 256) rem[w] = 0u;
    if (tid == 0) cnt = 0;
    __syncthreads();

    for (int i = 0; i < NTOP; ++i) {
        const bool alive = ((rem[i >> 5] >> (i & 31)) & 1u) == 0u;  // uniform read
        __syncthreads();                                            // reads before writes
        if (alive) {
            const unsigned* row = mask + (size_t)i * NWORD;
            for (int w = tid; w < NWORD; w += 256) rem[w] |= row[w];
            if (tid == 0 && cnt < NOUT) { keep[cnt] = i; cnt = cnt + 1; }
        }
        __syncthreads();
    }

    // keep = first 300 survivors, fill_value=0 -> box[0] for the tail
    for (int t = tid; t < NOUT; t += 256) {
        const int src = (t < cnt) ? keep[t] : 0;
        const float4 b = boxes[src];
        out[t * 5 + 0] = 0.0f;   // batch index
        out[t * 5 + 1] = b.x;
        out[t * 5 + 2] = b.y;
        out[t * 5 + 3] = b.z;
        out[t * 5 + 4] = b.w;
    }
}

// ---------------- Launcher ----------------
extern "C" void kernel_launch(void* const* d_in, const int* in_sizes, int n_in,
                              void* d_out, int out_size, void* d_ws, size_t ws_size,
                              hipStream_t stream) {
    const float* cls  = (const float*)d_in[0];   // (1,18,64,100) f32
    const float* bbox = (const float*)d_in[1];   // (1,36,64,100) f32
    const int*   ih   = (const int*)d_in[2];     // 1024
    const int*   iw   = (const int*)d_in[3];     // 1600

    // workspace layout (16B aligned slices): ~6.05 MB total
    char* ws = (char*)d_ws;
    float4*             proposals = (float4*)(ws + 0);                 // 57600*16 = 921600
    unsigned long long* keys      = (unsigned long long*)(ws + 921600);// 65536*8  = 524288
    float4*             boxes     = (float4*)(ws + 1445888);           // 6000*16  = 96000
    unsigned*           mask      = (unsigned*)(ws + 1541888);         // 6000*188*4 = 4512000

    float* out = (float*)d_out;                  // (300,5) f32

    // 1) decode + key pack (65536 threads)
    decode_kernel<<<SORT_N / 256, 256, 0, stream>>>(cls, bbox, ih, iw, proposals, keys);

    // 2) descending bitonic sort of 64K keys: hybrid LDS/global, 6 launches total.
    //    local_sort covers stages k=2..16384 in LDS; only 3 global passes remain.
    local_sort_kernel<<<SORT_N / CHUNK, SORT_THREADS, 0, stream>>>(keys);
    bitonic_pass_kernel<<<SORT_N / 256, 256, 0, stream>>>(keys, 32768, 16384);
    local_merge_kernel<<<SORT_N / CHUNK, SORT_THREADS, 0, stream>>>(keys, 32768);
    bitonic_pass_kernel<<<SORT_N / 256, 256, 0, stream>>>(keys, 65536, 32768);
    bitonic_pass_kernel<<<SORT_N / 256, 256, 0, stream>>>(keys, 65536, 16384);
    local_merge_kernel<<<SORT_N / CHUNK, SORT_THREADS, 0, stream>>>(keys, 65536);

    // 3) gather top-6000 boxes in score order
    gather_kernel<<<(NTOP + 255) / 256, 256, 0, stream>>>(keys, proposals, boxes);

    // 4) parallel suppression bitmask (async global->LDS staging)
    nms_mask_kernel<<<(NTOP + 63) / 64, 256, 0, stream>>>(boxes, mask);

    // 5) serial greedy sweep + output
    nms_sweep_kernel<<<1, 256, 0, stream>>>(boxes, mask, out);
}